// QEncoder_1451698946421
// MI455X (gfx1250) — compile-verified
//
#include <hip/hip_runtime.h>

typedef float v2f __attribute__((ext_vector_type(2)));
typedef float v8f __attribute__((ext_vector_type(8)));

#define NCODES 8192
#define CDIM   256
#define PLANE  4194304   /* 16*256*32*32 */
#define LDA    260       /* padded LDS row stride (dwords): spreads banks, keeps 8B align */

// ---------------------------------------------------------------------------
// Kernel 1: codebook squared norms  enorm[e] = sum_c emb[e][c]^2
// ---------------------------------------------------------------------------
__global__ void vq_enorm_kernel(const float* __restrict__ emb,
                                float* __restrict__ enorm) {
  int e = blockIdx.x * blockDim.x + threadIdx.x;
  if (e < NCODES) {
    const float4* r = (const float4*)(emb + (size_t)e * CDIM);
    float s = 0.f;
#pragma unroll 4
    for (int i = 0; i < CDIM / 4; ++i) {
      float4 v = r[i];
      s += v.x * v.x + v.y * v.y + v.z * v.z + v.w * v.w;
    }
    enorm[e] = s;
  }
}

// ---------------------------------------------------------------------------
// Kernel 2: per 16-pixel tile, argmin over all 8192 codes via f32 WMMA,
// then gather + write quantized / st / indices.
// ---------------------------------------------------------------------------
__global__ __launch_bounds__(256) void vq_argmin_kernel(
    const float* __restrict__ z, const float* __restrict__ emb,
    const float* __restrict__ enorm, float* __restrict__ out) {
  __shared__ float ldsA[16 * LDA];
  __shared__ unsigned long long best[16];
  __shared__ int bidx[16];

  const int tid   = threadIdx.x;
  const int mbase = blockIdx.x * 16;            // 16 consecutive pixels (same n,h)
  const int nimg  = mbase >> 10;
  const int h     = (mbase >> 5) & 31;
  const int w0    = mbase & 31;                 // 0 or 16
  const size_t zbase = (size_t)nimg * CDIM * 1024 + (size_t)h * 32 + w0;

  // Stage A tile: ldsA[m][c] = z_[mbase+m][c]  (global reads coalesced in w)
  for (int i = tid; i < 16 * CDIM; i += 256) {
    int m = i & 15, cc = i >> 4;
    ldsA[m * LDA + cc] = z[zbase + (size_t)cc * 1024 + m];
  }
  if (tid < 16) best[tid] = ~0ull;
  __syncthreads();

  const int wave  = tid >> 5;                   // 0..7
  const int lane  = tid & 31;
  const int laneN = lane & 15;                  // N (and A's M) index within tile
  const int koff  = (lane >> 4) << 1;           // K pair base: 0 (lanes 0-15) / 2 (16-31)
  const int rowOf = (lane >> 4) << 3;           // C/D rows: 0-7 (lo half) / 8-15 (hi half)

  float minval[8];
  int   minidx[8];
#pragma unroll
  for (int r = 0; r < 8; ++r) { minval[r] = 3.4e38f; minidx[r] = 0; }

  const float* aBase = &ldsA[laneN * LDA + koff];

  for (int t = 0; t < 64; ++t) {
    const int n = (t * 8 + wave) * 16 + laneN;  // waves interleaved for L2 locality
    const float* brow = emb + (size_t)n * CDIM + koff;
    v8f acc0 = {0.f, 0.f, 0.f, 0.f, 0.f, 0.f, 0.f, 0.f};
    v8f acc1 = {0.f, 0.f, 0.f, 0.f, 0.f, 0.f, 0.f, 0.f};
#pragma unroll 8
    for (int kt = 0; kt < 64; kt += 2) {        // K = 256 = 64 * 4, two indep. chains
      v2f a0 = *(const v2f*)(aBase + kt * 4);
      v2f b0 = *(const v2f*)(brow  + kt * 4);
      v2f a1 = *(const v2f*)(aBase + kt * 4 + 4);
      v2f b1 = *(const v2f*)(brow  + kt * 4 + 4);
      acc0 = __builtin_amdgcn_wmma_f32_16x16x4_f32(false, a0, false, b0,
                                                   (short)0, acc0, false, false);
      acc1 = __builtin_amdgcn_wmma_f32_16x16x4_f32(false, a1, false, b1,
                                                   (short)0, acc1, false, false);
    }
    const float en = enorm[n];
#pragma unroll
    for (int r = 0; r < 8; ++r) {
      float s = en - 2.f * (acc0[r] + acc1[r]); // ||e||^2 - 2 z.e  (argmin-equivalent)
      if (s < minval[r]) { minval[r] = s; minidx[r] = n; }
    }
  }

  // Reduce across the 16 N-columns held by each half-wave.
#pragma unroll
  for (int off = 8; off >= 1; off >>= 1) {
#pragma unroll
    for (int r = 0; r < 8; ++r) {
      float ov = __shfl_xor(minval[r], off, 32);
      int   oi = __shfl_xor(minidx[r], off, 32);
      if (ov < minval[r] || (ov == minval[r] && oi < minidx[r])) {
        minval[r] = ov; minidx[r] = oi;
      }
    }
  }
  // Cross-wave merge: packed (monotone-float, index) u64 min in LDS.
  if (laneN == 0) {
#pragma unroll
    for (int r = 0; r < 8; ++r) {
      unsigned int u = __float_as_uint(minval[r]);
      u = (u & 0x80000000u) ? ~u : (u | 0x80000000u);
      unsigned long long key =
          ((unsigned long long)u << 32) | (unsigned int)minidx[r];
      atomicMin(&best[rowOf + r], key);
    }
  }
  __syncthreads();

  if (tid < 16) {
    int idx = (int)(unsigned int)(best[tid] & 0xffffffffull);
    bidx[tid] = idx;
    out[2 * (size_t)PLANE + mbase + tid] = (float)idx;   // encoding_indices
  }
  __syncthreads();

  // Gather winners, write quantized and st (identical values) in NCHW.
  for (int i = tid; i < 16 * CDIM; i += 256) {
    int m = i & 15, cc = i >> 4;
    float v = emb[(size_t)bidx[m] * CDIM + cc];
    size_t addr = zbase + (size_t)cc * 1024 + m;
    out[addr] = v;
    out[(size_t)PLANE + addr] = v;
  }
}

// ---------------------------------------------------------------------------
extern "C" void kernel_launch(void* const* d_in, const int* in_sizes, int n_in,
                              void* d_out, int out_size, void* d_ws, size_t ws_size,
                              hipStream_t stream) {
  const float* z   = (const float*)d_in[0];   // (16, 256, 32, 32) fp32
  const float* emb = (const float*)d_in[1];   // (8192, 256) fp32
  float* out   = (float*)d_out;               // quantized | st | indices
  float* enorm = (float*)d_ws;                // 8192 floats of scratch

  vq_enorm_kernel<<<NCODES / 256, 256, 0, stream>>>(emb, enorm);
  vq_argmin_kernel<<<16384 / 16, 256, 0, stream>>>(z, emb, enorm, out);
}